// GridSampler3D_17093969838614
// MI455X (gfx1250) — compile-verified
//
#include <hip/hip_runtime.h>
#include <hip/hip_bf16.h>

#ifndef __has_builtin
#define __has_builtin(x) 0
#endif

// 4-byte-aligned float2 so the compiler may emit a single global_load_b64 for
// the (x0, x0+1) corner pair even when x0 is odd (UNALIGNED mem mode).
typedef float v2f __attribute__((ext_vector_type(2), aligned(4)));

namespace {
constexpr int kN = 8, kC = 32, kD = 64, kH = 64, kW = 64;
constexpr int kDHW     = kD * kH * kW;   // per-channel volume, 262144
constexpr int kSpatial = 64 * 64 * 64;   // output points per batch
constexpr int kBlock   = 256;            // 8 waves (wave32) per workgroup

__device__ __forceinline__ int clampi(int v, int lo, int hi) {
  return v < lo ? lo : (v > hi ? hi : v);
}
} // namespace

__global__ __launch_bounds__(kBlock)
void grid_sample3d_kernel(const float* __restrict__ vol,
                          const float* __restrict__ grid,
                          float* __restrict__ out)
{
  __shared__ float sg[kBlock * 3];           // staged (x,y,z) per point
  const int t = threadIdx.x;
  const long long p0 = (long long)blockIdx.x * kBlock;

  // ---- Stage this block's grid coords via CDNA5 async global->LDS DMA ----
  const float* gsrc = grid + p0 * 3;         // 3072 contiguous bytes
#if __has_builtin(__builtin_amdgcn_global_load_async_to_lds_b32)
  {
    // Builtin prototype takes int* operands (global src, LDS dst).
    auto g = (__attribute__((address_space(1))) int*)(const_cast<float*>(gsrc) + t);
    auto l = (__attribute__((address_space(3))) int*)(sg + t);
    // IOFFSET applies to both the global and LDS address (ISA 15.18.3).
    __builtin_amdgcn_global_load_async_to_lds_b32(g, l, 0, 0);
    __builtin_amdgcn_global_load_async_to_lds_b32(g, l, kBlock * 4, 0);
    __builtin_amdgcn_global_load_async_to_lds_b32(g, l, 2 * kBlock * 4, 0);
  }
#if __has_builtin(__builtin_amdgcn_s_wait_asynccnt)
  __builtin_amdgcn_s_wait_asynccnt(0);
#else
  asm volatile("s_wait_asynccnt 0" ::: "memory");
#endif
#else
  sg[t]              = gsrc[t];
  sg[t + kBlock]     = gsrc[t + kBlock];
  sg[t + 2 * kBlock] = gsrc[t + 2 * kBlock];
#endif
  __syncthreads();

  // ---- Per-point sample setup ----
  const long long p  = p0 + t;
  const int n  = (int)(p >> 18);               // / 262144
  const int sp = (int)(p & (kSpatial - 1));

  const float gx = sg[3 * t + 0];
  const float gy = sg[3 * t + 1];
  const float gz = sg[3 * t + 2];

  // align_corners=True: [-1,1] -> [0, size-1]
  const float ix = (gx + 1.0f) * (0.5f * (kW - 1));
  const float iy = (gy + 1.0f) * (0.5f * (kH - 1));
  const float iz = (gz + 1.0f) * (0.5f * (kD - 1));

  // Clamp low corner to [0, size-2]; frac compensates so ix==size-1 gives
  // fx==1 -> exact endpoint. Identical to reference for grid in [-1,1].
  const int x0 = clampi((int)floorf(ix), 0, kW - 2);
  const int y0 = clampi((int)floorf(iy), 0, kH - 2);
  const int z0 = clampi((int)floorf(iz), 0, kD - 2);
  const float fx = ix - (float)x0;
  const float fy = iy - (float)y0;
  const float fz = iz - (float)z0;

  const float* vbase = vol + (long long)n * kC * kDHW
                           + ((z0 << 12) + (y0 << 6) + x0);
  float* obase = out + (long long)n * kC * kSpatial + sp;

  // ---- 32 channels: 4 x b64 pair-gathers + 7 lerps each ----
#pragma unroll 4
  for (int c = 0; c < kC; ++c) {
    const float* vp = vbase + (long long)c * kDHW;
    const v2f v00 = *(const v2f*)(vp);                 // (z0,y0,x0..x1)
    const v2f v01 = *(const v2f*)(vp + kW);            // (z0,y1)
    const v2f v10 = *(const v2f*)(vp + kH * kW);       // (z1,y0)
    const v2f v11 = *(const v2f*)(vp + kH * kW + kW);  // (z1,y1)

    const float a  = fmaf(fx, v00.y - v00.x, v00.x);
    const float b  = fmaf(fx, v01.y - v01.x, v01.x);
    const float cc = fmaf(fx, v10.y - v10.x, v10.x);
    const float d  = fmaf(fx, v11.y - v11.x, v11.x);
    const float e  = fmaf(fy, b - a, a);
    const float f  = fmaf(fy, d - cc, cc);
    const float r  = fmaf(fz, f - e, e);

    // Streaming store (TH_NT): keep the 256 MiB volume resident in L2.
    __builtin_nontemporal_store(r, obase + (long long)c * kSpatial);
  }
}

extern "C" void kernel_launch(void* const* d_in, const int* in_sizes, int n_in,
                              void* d_out, int out_size, void* d_ws, size_t ws_size,
                              hipStream_t stream) {
  const float* vol  = (const float*)d_in[0];  // [8,32,64,64,64] f32
  const float* grd  = (const float*)d_in[1];  // [8,64,64,64,3]  f32
  float* out = (float*)d_out;                 // [8,32,64,64,64] f32

  const long long points = (long long)kN * kSpatial;        // 2,097,152
  const unsigned blocks  = (unsigned)(points / kBlock);     // 8192
  grid_sample3d_kernel<<<blocks, kBlock, 0, stream>>>(vol, grd, out);
}